// SegmentClassifier_89644557402740
// MI455X (gfx1250) — compile-verified
//
#include <hip/hip_runtime.h>
#include <math.h>

typedef __attribute__((ext_vector_type(2))) float v2f;
typedef __attribute__((ext_vector_type(8))) float v8f;

#define LN_EPS 1e-5f

struct BlockP { const float *W, *b, *g, *be; };

// ---------------- scalar MLP block: out8 = tanh(LN(x@W + b)*g + be) ----------
template <int DIN>
__device__ __forceinline__ void mlp_block(const BlockP p, const float* x, float* out8) {
    float y[8];
#pragma unroll
    for (int j = 0; j < 8; ++j) y[j] = p.b[j];
#pragma unroll
    for (int k = 0; k < DIN; ++k) {
        const float xk = x[k];
#pragma unroll
        for (int j = 0; j < 8; ++j) y[j] = fmaf(xk, p.W[k * 8 + j], y[j]);
    }
    float mu = 0.f;
#pragma unroll
    for (int j = 0; j < 8; ++j) mu += y[j];
    mu *= 0.125f;
    float var = 0.f;
#pragma unroll
    for (int j = 0; j < 8; ++j) { const float d = y[j] - mu; var += d * d; }
    const float inv = rsqrtf(var * 0.125f + LN_EPS);
#pragma unroll
    for (int j = 0; j < 8; ++j) out8[j] = tanhf((y[j] - mu) * inv * p.g[j] + p.be[j]);
}

// ---------------- input net: h = MLP(x); H = [2h, x, pad] ---------------------
__global__ void __launch_bounds__(256)
input_net_kernel(const float* __restrict__ nodes, BlockP p0, BlockP p1, BlockP p2,
                 float* __restrict__ H, float* __restrict__ hbuf, int N) {
    const int n = blockIdx.x * blockDim.x + threadIdx.x;
    if (n >= N) return;
    float x[3];
#pragma unroll
    for (int k = 0; k < 3; ++k) x[k] = nodes[(size_t)n * 3 + k];
    float t1[8], t2[8], h[8];
    mlp_block<3>(p0, x, t1);
    mlp_block<8>(p1, t1, t2);
    mlp_block<8>(p2, t2, h);
    float* Hr = H + (size_t)n * 12;
#pragma unroll
    for (int j = 0; j < 8; ++j) Hr[j] = 2.f * h[j];
#pragma unroll
    for (int k = 0; k < 3; ++k) Hr[8 + k] = x[k];
    Hr[11] = 0.f;
#pragma unroll
    for (int j = 0; j < 8; ++j) hbuf[(size_t)n * 8 + j] = h[j];
}

// ---------------- node net: h=MLP([m_in,m_out,H]); H[:8]=h0+h -----------------
__global__ void __launch_bounds__(256)
node_net_kernel(const float* __restrict__ m_in, const float* __restrict__ m_out,
                BlockP p0, BlockP p1, BlockP p2,
                float* __restrict__ H, float* __restrict__ hbuf, int N) {
    const int n = blockIdx.x * blockDim.x + threadIdx.x;
    if (n >= N) return;
    float M[33];
    const float* mi = m_in + (size_t)n * 12;
    const float* mo = m_out + (size_t)n * 12;
    float* Hr = H + (size_t)n * 12;
#pragma unroll
    for (int k = 0; k < 11; ++k) { M[k] = mi[k]; M[11 + k] = mo[k]; M[22 + k] = Hr[k]; }
    float t1[8], t2[8], hn[8];
    mlp_block<33>(p0, M, t1);
    mlp_block<8>(p1, t1, t2);
    mlp_block<8>(p2, t2, hn);
#pragma unroll
    for (int j = 0; j < 8; ++j) {
        const float h0 = hbuf[(size_t)n * 8 + j];
        Hr[j] = h0 + hn[j];
        hbuf[(size_t)n * 8 + j] = hn[j];
    }
    // Hr[8..10] (= x) unchanged
}

// ---------------- edge net via f32 WMMA (16 edges / wave tile) ----------------
// Per-wave LDS staging: sX = 16 edges x 24 padded input features,
//                       sA = 16 edges x 8 layer-1 activations.
__global__ void __launch_bounds__(256)
edge_net_kernel(const int* __restrict__ edges, const float* __restrict__ H,
                BlockP p1, BlockP p2, const float* __restrict__ Wf,
                const float* __restrict__ bfp,
                float* __restrict__ m_in, float* __restrict__ m_out,
                float* __restrict__ out_w, int E, int do_scatter) {
    __shared__ __align__(16) float sX[8][16 * 24];
    __shared__ __align__(16) float sA[8][16 * 8];

    const int lane = threadIdx.x & 31;
    const int col  = lane & 15;     // edge slot within tile / output-feature column
    const int half = lane >> 4;     // K-half selector per f32 WMMA layout
    const int wib  = threadIdx.x >> 5;
    const int wavesPerBlock = blockDim.x >> 5;
    const int wave  = blockIdx.x * wavesPerBlock + wib;
    const int nwave = gridDim.x * wavesPerBlock;
    const int ntiles = (E + 15) >> 4;

    // A fragments: A = W^T (M=out feature=col, K=in feature), K padded to mult of 4
    v2f a1[6], a2[2];
#pragma unroll
    for (int c = 0; c < 6; ++c) {
        const int k0 = 4 * c + 2 * half;
        float lo = 0.f, hi = 0.f;
        if (col < 8) {
            if (k0 < 22)     lo = p1.W[k0 * 8 + col];
            if (k0 + 1 < 22) hi = p1.W[(k0 + 1) * 8 + col];
        }
        a1[c].x = lo; a1[c].y = hi;
    }
#pragma unroll
    for (int c = 0; c < 2; ++c) {
        const int k0 = 4 * c + 2 * half;
        a2[c].x = (col < 8) ? p2.W[k0 * 8 + col] : 0.f;
        a2[c].y = (col < 8) ? p2.W[(k0 + 1) * 8 + col] : 0.f;
    }
    float b1v[8], g1v[8], be1v[8], b2v[8], g2v[8], be2v[8], wfv[8];
#pragma unroll
    for (int j = 0; j < 8; ++j) {
        b1v[j] = p1.b[j]; g1v[j] = p1.g[j]; be1v[j] = p1.be[j];
        b2v[j] = p2.b[j]; g2v[j] = p2.g[j]; be2v[j] = p2.be[j];
        wfv[j] = Wf[j];
    }
    const float bf0 = bfp[0];

    for (int t = wave; t < ntiles; t += nwave) {   // wave-uniform loop (EXEC all-1s)
        const int e = t * 16 + col;
        const bool valid = (e < E);
        const int ec = valid ? e : (E - 1);
        const int s = edges[2 * ec + 0];
        const int d = edges[2 * ec + 1];

        // each half-wave gathers only its own row: lower = H[src], upper = H[dst]
        const float* rowp = half ? (H + (size_t)d * 12) : (H + (size_t)s * 12);
        float r[11];
#pragma unroll
        for (int k = 0; k < 11; ++k) r[k] = rowp[k];

        // stage the 16x24 tile image into per-wave LDS (lane writes its own slot)
        float* xrow = &sX[wib][col * 24 + 11 * half];
#pragma unroll
        for (int k = 0; k < 11; ++k) xrow[k] = r[k];
        if (half) {                         // K-pad (elements 22,23) = 0
            sX[wib][col * 24 + 22] = 0.f;
            sX[wib][col * 24 + 23] = 0.f;
        }
        // same-wave LDS ops are in-order; no barrier needed (per-wave region)

        // ---- layer 1: D = W1^T (16x24) x X^T (24x16), 6 chained WMMAs ----
        v8f acc = {0.f, 0.f, 0.f, 0.f, 0.f, 0.f, 0.f, 0.f};
#pragma unroll
        for (int c = 0; c < 6; ++c) {
            const v2f bf_ = *(const v2f*)&sX[wib][col * 24 + 4 * c + 2 * half];
            acc = __builtin_amdgcn_wmma_f32_16x16x4_f32(false, a1[c], false, bf_,
                                                        (short)0, acc, false, false);
        }
        // bias + LN + tanh — lane-local (lane col<16 holds all 8 features of its edge)
        float act[8];
        {
            float y[8]; float mu = 0.f;
#pragma unroll
            for (int j = 0; j < 8; ++j) { y[j] = acc[j] + b1v[j]; mu += y[j]; }
            mu *= 0.125f;
            float var = 0.f;
#pragma unroll
            for (int j = 0; j < 8; ++j) { const float dd = y[j] - mu; var += dd * dd; }
            const float inv = rsqrtf(var * 0.125f + LN_EPS);
#pragma unroll
            for (int j = 0; j < 8; ++j) act[j] = tanhf((y[j] - mu) * inv * g1v[j] + be1v[j]);
        }
        // stage activations: lower half owns valid rows (out-features 0..7)
        if (half == 0) {
#pragma unroll
            for (int j = 0; j < 8; ++j) sA[wib][col * 8 + j] = act[j];
        }

        // ---- layer 2: D = W2^T (16x8) x act^T (8x16), 2 WMMAs ----
        v8f acc2 = {0.f, 0.f, 0.f, 0.f, 0.f, 0.f, 0.f, 0.f};
#pragma unroll
        for (int c = 0; c < 2; ++c) {
            const v2f bf_ = *(const v2f*)&sA[wib][col * 8 + 4 * c + 2 * half];
            acc2 = __builtin_amdgcn_wmma_f32_16x16x4_f32(false, a2[c], false, bf_,
                                                         (short)0, acc2, false, false);
        }
        float act2[8];
        {
            float y[8]; float mu = 0.f;
#pragma unroll
            for (int j = 0; j < 8; ++j) { y[j] = acc2[j] + b2v[j]; mu += y[j]; }
            mu *= 0.125f;
            float var = 0.f;
#pragma unroll
            for (int j = 0; j < 8; ++j) { const float dd = y[j] - mu; var += dd * dd; }
            const float inv = rsqrtf(var * 0.125f + LN_EPS);
#pragma unroll
            for (int j = 0; j < 8; ++j) act2[j] = tanhf((y[j] - mu) * inv * g2v[j] + be2v[j]);
        }
        // ---- final 8->1 + sigmoid (lane-local on lower half) ----
        float z = bf0;
#pragma unroll
        for (int j = 0; j < 8; ++j) z = fmaf(act2[j], wfv[j], z);
        float w = 1.f / (1.f + __expf(-z));
        w = __shfl(w, col, 32);             // both halves get their edge's weight

        if (do_scatter) {
            if (valid) {
                // unified scatter: lower half -> m_in[dst] += w*H[src]
                //                  upper half -> m_out[src] += w*H[dst]
                float* tgt = half ? (m_out + (size_t)s * 12) : (m_in + (size_t)d * 12);
#pragma unroll
                for (int k = 0; k < 11; ++k) atomicAdd(&tgt[k], w * r[k]);
            }
        } else {
            if (valid && half == 0) out_w[e] = w;
        }
    }
}

// -----------------------------------------------------------------------------
extern "C" void kernel_launch(void* const* d_in, const int* in_sizes, int n_in,
                              void* d_out, int out_size, void* d_ws, size_t ws_size,
                              hipStream_t stream) {
    const float* nodes = (const float*)d_in[0];
    const int*   edges = (const int*)d_in[1];
    const int N = in_sizes[0] / 3;
    const int E = in_sizes[1] / 2;

    auto blk = [&](int base) {
        BlockP p;
        p.W  = (const float*)d_in[base + 0];
        p.b  = (const float*)d_in[base + 1];
        p.g  = (const float*)d_in[base + 2];
        p.be = (const float*)d_in[base + 3];
        return p;
    };
    // input_params: d_in[2..13], edge_params: d_in[14..23], node_params: d_in[24..35]
    BlockP ip0 = blk(2),  ip1 = blk(6),  ip2 = blk(10);
    BlockP ep0 = blk(14), ep1 = blk(18);
    const float* Wf = (const float*)d_in[22];
    const float* bf = (const float*)d_in[23];
    BlockP np0 = blk(24), np1 = blk(28), np2 = blk(32);

    float* H     = (float*)d_ws;                    // N x 12
    float* hbuf  = H     + (size_t)N * 12;          // N x 8
    float* m_in  = hbuf  + (size_t)N * 8;           // N x 12
    float* m_out = m_in  + (size_t)N * 12;          // N x 12
    float* out_w = (float*)d_out;                   // E

    const int TB = 256;
    const int nodeBlocks = (N + TB - 1) / TB;
    const int ntiles = (E + 15) / 16;
    const int edgeBlocks = (ntiles + 7) / 8;        // 8 waves of 32 per block

    input_net_kernel<<<nodeBlocks, TB, 0, stream>>>(nodes, ip0, ip1, ip2, H, hbuf, N);

    for (int it = 0; it < 3; ++it) {
        hipMemsetAsync(m_in,  0, (size_t)N * 12 * sizeof(float), stream);
        hipMemsetAsync(m_out, 0, (size_t)N * 12 * sizeof(float), stream);
        edge_net_kernel<<<edgeBlocks, TB, 0, stream>>>(edges, H, ep0, ep1, Wf, bf,
                                                       m_in, m_out, out_w, E, 1);
        node_net_kernel<<<nodeBlocks, TB, 0, stream>>>(m_in, m_out, np0, np1, np2,
                                                       H, hbuf, N);
    }
    edge_net_kernel<<<edgeBlocks, TB, 0, stream>>>(edges, H, ep0, ep1, Wf, bf,
                                                   m_in, m_out, out_w, E, 0);
}